// SpikeFP32Embedding_43860206027345
// MI455X (gfx1250) — compile-verified
//
#include <hip/hip_runtime.h>
#include <stdint.h>

// ---------------------------------------------------------------------------
// SpikeFP32Embedding on MI455X (gfx1250): pure HBM-bound row gather.
//   table:  [32768, 128, 32] f32  (row = 16 KB)
//   tokens: 8*2048 = 16384 ids
//   out:    [16384, 128, 32] f32  (256 MB)
// Roofline: 256 MB read + 256 MB write => ~22 us @ 23.3 TB/s. Zero FLOPs, so
// the right CDNA5 path is the Tensor Data Mover's *gather mode*: one
// TENSOR_LOAD_TO_LDS gathers N table rows (16-bit row indices in D# group2),
// one TENSOR_STORE_FROM_LDS streams them to the output. No VGPR round-trip.
// Toolchain probe: this clang wants the 6-arg builtin form
//   (uint32x4 g0, int32x8 g1, int32x4 g2, int32x4 g3, int32x8, i32 cpol).
// ---------------------------------------------------------------------------

#define EMBED       128
#define PULSE       32
#define ROW_FLOATS  (EMBED * PULSE)      // 4096 floats
#define ROW_BYTES   (ROW_FLOATS * 4)     // 16384 B per row
#define ROW_UNITS8  (ROW_BYTES / 8)      // 2048 8-byte units (data_size=3)
#define TOK_PER_WG  2
#define LDS_BYTES   (TOK_PER_WG * ROW_BYTES)   // 32 KB staged in LDS

#if defined(__gfx1250__) && __has_builtin(__builtin_amdgcn_tensor_load_to_lds) \
                         && __has_builtin(__builtin_amdgcn_tensor_store_from_lds)
#define USE_TDM 1
#else
#define USE_TDM 0
#endif

typedef unsigned int v4u __attribute__((ext_vector_type(4)));
typedef int          v8i __attribute__((ext_vector_type(8)));
typedef int          v4i __attribute__((ext_vector_type(4)));

#if USE_TDM
// One wave per workgroup; each workgroup gathers TOK_PER_WG rows via TDM.
__global__ __launch_bounds__(32)
void SpikeFP32Embedding_tdm_kernel(const int* __restrict__ token_ids,
                                   const float* __restrict__ weight,
                                   float* __restrict__ out)
{
    __shared__ __align__(128) char smem[LDS_BYTES];
    // AS3 pointer through flat aperture: low 32 bits == LDS byte offset.
    uint32_t lds_off = (uint32_t)(uintptr_t)&smem[0];

    int base = (int)blockIdx.x * TOK_PER_WG;

    // Uniform-address loads; force SGPR for descriptor packing.
    uint32_t id0 = (uint32_t)token_ids[base + 0];
    uint32_t id1 = (uint32_t)token_ids[base + 1];
    id0 = (uint32_t)__builtin_amdgcn_readfirstlane((int)id0);
    id1 = (uint32_t)__builtin_amdgcn_readfirstlane((int)id1);

    // ---------------- Gather load: weight rows -> LDS ----------------
    uint64_t wbase = (uint64_t)(uintptr_t)weight;
    v4u g0 = {
        0x80000001u,                                   // count=1, gather_mode=1, 16-bit idx
        lds_off,                                       // lds_addr
        (uint32_t)(wbase & 0xFFFFFFFFu),               // global_addr[31:0]
        ((uint32_t)((wbase >> 32) & 0x01FFFFFFu))      // global_addr[56:32]
            | 0x80000000u                              // type=2 ("image") in bits 127:126
    };
    v8i g1 = {
        (int)(3u << 16),                               // workgroup_mask=0, data_size=3 (8B)
        (int)((uint32_t)ROW_UNITS8 << 16),             // tensor_dim0[15:0] in bits 63:48
        (int)(0x8000u << 16),                          // tensor_dim0[31:16]=0; tensor_dim1=32768 (lo16)
        (int)((uint32_t)ROW_UNITS8 << 16),             // tensor_dim1[31:16]=0; tile_dim0=2048
        (int)TOK_PER_WG,                               // tile_dim1 = #valid gather indices
        (int)ROW_UNITS8,                               // tensor_dim0_stride[31:0] = 2048 units
        0, 0                                           // stride hi / dim1_stride (ignored in gather)
    };
    v4i g2 = { (int)(id0 | (id1 << 16)), 0, 0, 0 };    // row_index_0, row_index_1 (16-bit mode)
    v4i g3 = { 0, 0, 0, 0 };
    v8i xtra = { 0, 0, 0, 0, 0, 0, 0, 0 };             // 6-arg form: extra operand, zero-filled
    __builtin_amdgcn_tensor_load_to_lds(g0, g1, g2, g3, xtra, /*cpol=*/0);

    // TDM done signal -> TENSORcnt; wait until the 32 KB tile is in LDS.
    __builtin_amdgcn_s_wait_tensorcnt(0);

    // ---------------- Linear store: LDS -> out (32 KB contiguous) ----------------
    uint64_t obase = (uint64_t)(uintptr_t)out + (uint64_t)base * ROW_BYTES;
    const uint32_t STORE_UNITS = TOK_PER_WG * ROW_UNITS8;   // 4096 8-byte units
    v4u s0 = {
        0x00000001u,                                   // count=1, normal mode
        lds_off,
        (uint32_t)(obase & 0xFFFFFFFFu),
        ((uint32_t)((obase >> 32) & 0x01FFFFFFu)) | 0x80000000u
    };
    v8i s1 = {
        (int)(3u << 16),                               // data_size=3 (8B)
        (int)((STORE_UNITS & 0xFFFFu) << 16),          // tensor_dim0 lo16 = 4096
        (int)(1u << 16),                               // tensor_dim0 hi16=0; tensor_dim1=1
        (int)((STORE_UNITS & 0xFFFFu) << 16),          // tile_dim0 = 4096 (1-D tile)
        0,                                             // tile_dim1=0 (unused), tile_dim2=0
        (int)STORE_UNITS,                              // tensor_dim0_stride
        0, 0
    };
    v4i z = { 0, 0, 0, 0 };
    __builtin_amdgcn_tensor_store_from_lds(s0, s1, z, z, xtra, /*cpol=*/0);

    // S_ENDPGM implies wait-idle, but make the dependency explicit.
    __builtin_amdgcn_s_wait_tensorcnt(0);
}
#endif // USE_TDM

// Fallback: plain wide-vector streaming gather (still HBM-roofline optimal).
__global__ __launch_bounds__(256)
void SpikeFP32Embedding_copy_kernel(const int* __restrict__ token_ids,
                                    const float4* __restrict__ weight,
                                    float4* __restrict__ out)
{
    int tok = (int)blockIdx.x;
    int row = token_ids[tok];
    const float4* src = weight + (size_t)row * (ROW_FLOATS / 4);
    float4*       dst = out    + (size_t)tok * (ROW_FLOATS / 4);
    #pragma unroll 4
    for (int i = (int)threadIdx.x; i < ROW_FLOATS / 4; i += 256)
        dst[i] = src[i];
}

extern "C" void kernel_launch(void* const* d_in, const int* in_sizes, int n_in,
                              void* d_out, int out_size, void* d_ws, size_t ws_size,
                              hipStream_t stream)
{
    (void)n_in; (void)out_size; (void)d_ws; (void)ws_size;
    const int*   token_ids = (const int*)d_in[0];     // harness delivers integer inputs as int32
    const float* weight    = (const float*)d_in[1];   // [32768, 128, 32] f32 bit-planes
    float*       out       = (float*)d_out;           // [16384, 128, 32] f32

    int n_tok = in_sizes[0];                          // 8 * 2048 = 16384

#if USE_TDM
    int n_wg = n_tok / TOK_PER_WG;                    // 8192 workgroups, 1 wave each
    hipLaunchKernelGGL(SpikeFP32Embedding_tdm_kernel,
                       dim3(n_wg), dim3(32), 0, stream,
                       token_ids, weight, out);
#else
    hipLaunchKernelGGL(SpikeFP32Embedding_copy_kernel,
                       dim3(n_tok), dim3(256), 0, stream,
                       token_ids, (const float4*)weight, (float4*)out);
#endif
}